// GaussianModel_64725157151170
// MI455X (gfx1250) — compile-verified
//
#include <hip/hip_runtime.h>
#include <cstdint>
#include <cstddef>

#define TPB 256
#define PPB 256  // points per block

__device__ __forceinline__ uint32_t lds_addr_of(const void* p) {
  // Generic pointer to LDS: low 32 bits are the wave-relative LDS byte offset.
  return (uint32_t)(uintptr_t)p;
}

__global__ __launch_bounds__(TPB) void cov3d_kernel(
    const float* __restrict__ scales,   // N x 3
    const float* __restrict__ rots,     // N x 4
    float* __restrict__ out,            // N x 3 x 3
    int n)
{
  __shared__ float sS[PPB * 3];   // 3072 B staged scales
  __shared__ float sR[PPB * 4];   // 4096 B staged rotations
  __shared__ float sC[PPB * 9];   // 9216 B staged covariances

  const int tid  = threadIdx.x;
  const int base = blockIdx.x * PPB;
  int rem = n - base;
  const int npts = rem < PPB ? rem : PPB;
  const bool full = (npts == PPB);

  float a0 = 0.f, a1 = 0.f, a2 = 0.f;
  float w = 0.f, x = 0.f, y = 0.f, z = 0.f;

  if (full) {
    // ---- async global -> LDS staging, 16B per lane-transfer, non-temporal ----
    const uint64_t gS = (uint64_t)(uintptr_t)scales + (uint64_t)base * 12u;
    const uint64_t gR = (uint64_t)(uintptr_t)rots   + (uint64_t)base * 16u;
    const uint32_t lS = lds_addr_of(sS);
    const uint32_t lR = lds_addr_of(sR);

    if (tid < (PPB * 3 * 4) / 16) {          // 192 transfers for scales
      uint64_t ga = gS + (uint64_t)tid * 16u;
      uint32_t la = lS + (uint32_t)tid * 16u;
      asm volatile("global_load_async_to_lds_b128 %0, %1, off th:TH_LOAD_NT"
                   :: "v"(la), "v"(ga) : "memory");
    }
    {                                        // 256 transfers for rotations
      uint64_t ga = gR + (uint64_t)tid * 16u;
      uint32_t la = lR + (uint32_t)tid * 16u;
      asm volatile("global_load_async_to_lds_b128 %0, %1, off th:TH_LOAD_NT"
                   :: "v"(la), "v"(ga) : "memory");
    }
    asm volatile("s_wait_asynccnt 0" ::: "memory");
    __syncthreads();

    a0 = sS[tid * 3 + 0];
    a1 = sS[tid * 3 + 1];
    a2 = sS[tid * 3 + 2];
    const float4 q4 = *(const float4*)&sR[tid * 4];
    w = q4.x; x = q4.y; y = q4.z; z = q4.w;
  } else if (tid < npts) {
    const int i = base + tid;
    a0 = scales[i * 3 + 0];
    a1 = scales[i * 3 + 1];
    a2 = scales[i * 3 + 2];
    w = rots[i * 4 + 0];
    x = rots[i * 4 + 1];
    y = rots[i * 4 + 2];
    z = rots[i * 4 + 3];
  }

  float c00 = 0.f, c01 = 0.f, c02 = 0.f, c11 = 0.f, c12 = 0.f, c22 = 0.f;
  if (full || tid < npts) {
    // s^2 = exp(clip(scales, -10, 2))^2
    float t0 = fminf(fmaxf(a0, -10.f), 2.f);
    float t1 = fminf(fmaxf(a1, -10.f), 2.f);
    float t2 = fminf(fmaxf(a2, -10.f), 2.f);
    float sx = expf(t0), sy = expf(t1), sz = expf(t2);
    float s2x = sx * sx, s2y = sy * sy, s2z = sz * sz;

    // first normalization: q / (||q|| + 1e-8)
    float nr  = sqrtf(w * w + x * x + y * y + z * z);
    float inv = 1.0f / (nr + 1e-8f);
    w *= inv; x *= inv; y *= inv; z *= inv;
    // second normalization (reference normalizes again inside _quat_to_rotmat)
    float nr2  = sqrtf(w * w + x * x + y * y + z * z);
    float inv2 = 1.0f / nr2;
    w *= inv2; x *= inv2; y *= inv2; z *= inv2;

    float xx = x * x, yy = y * y, zz = z * z;
    float xy = x * y, xz = x * z, yz = y * z;
    float wx = w * x, wy = w * y, wz = w * z;

    float r00 = 1.f - 2.f * (yy + zz);
    float r01 = 2.f * (xy - wz);
    float r02 = 2.f * (xz + wy);
    float r10 = 2.f * (xy + wz);
    float r11 = 1.f - 2.f * (xx + zz);
    float r12 = 2.f * (yz - wx);
    float r20 = 2.f * (xz - wy);
    float r21 = 2.f * (yz + wx);
    float r22 = 1.f - 2.f * (xx + yy);

    // M = R * diag(s^2); cov = M * R^T (symmetric, compute 6 entries)
    float m00 = r00 * s2x, m01 = r01 * s2y, m02 = r02 * s2z;
    float m10 = r10 * s2x, m11 = r11 * s2y, m12 = r12 * s2z;
    float m20 = r20 * s2x, m21 = r21 * s2y, m22 = r22 * s2z;

    c00 = m00 * r00 + m01 * r01 + m02 * r02;
    c01 = m00 * r10 + m01 * r11 + m02 * r12;
    c02 = m00 * r20 + m01 * r21 + m02 * r22;
    c11 = m10 * r10 + m11 * r11 + m12 * r12;
    c12 = m10 * r20 + m11 * r21 + m12 * r22;
    c22 = m20 * r20 + m21 * r21 + m22 * r22;
  }

  if (full) {
    // stage into LDS (stride-9 dwords across lanes: conflict-free on 64 banks)
    float* cv = &sC[tid * 9];
    cv[0] = c00; cv[1] = c01; cv[2] = c02;
    cv[3] = c01; cv[4] = c11; cv[5] = c12;
    cv[6] = c02; cv[7] = c12; cv[8] = c22;
    __syncthreads();

    // ---- async LDS -> global streaming store, 576 x 16B per block, non-temporal ----
    const uint64_t gC = (uint64_t)(uintptr_t)out + (uint64_t)base * 36u;
    const uint32_t lC = lds_addr_of(sC);
    const int xfers = (PPB * 9 * 4) / 16;  // 576
#pragma unroll
    for (int u = 0; u < xfers; u += TPB) {
      int idx = u + tid;
      if (idx < xfers) {
        uint64_t ga = gC + (uint64_t)idx * 16u;
        uint32_t la = lC + (uint32_t)idx * 16u;
        asm volatile("global_store_async_from_lds_b128 %0, %1, off th:TH_STORE_NT"
                     :: "v"(ga), "v"(la) : "memory");
      }
    }
    asm volatile("s_wait_asynccnt 0" ::: "memory");
  } else if (tid < npts) {
    float* o = out + (size_t)(base + tid) * 9;
    o[0] = c00; o[1] = c01; o[2] = c02;
    o[3] = c01; o[4] = c11; o[5] = c12;
    o[6] = c02; o[7] = c12; o[8] = c22;
  }
}

extern "C" void kernel_launch(void* const* d_in, const int* in_sizes, int n_in,
                              void* d_out, int out_size, void* d_ws, size_t ws_size,
                              hipStream_t stream) {
  (void)n_in; (void)out_size; (void)d_ws; (void)ws_size;
  const float* scales = (const float*)d_in[0];
  const float* rots   = (const float*)d_in[1];
  float* out = (float*)d_out;
  const int n = in_sizes[0] / 3;
  if (n <= 0) return;
  const int blocks = (n + PPB - 1) / PPB;
  cov3d_kernel<<<blocks, TPB, 0, stream>>>(scales, rots, out, n);
}